// CSMAdapter_30227979829783
// MI455X (gfx1250) — compile-verified
//
#include <hip/hip_runtime.h>
#include <hip/hip_bf16.h>
#include <math.h>

// MI455X / gfx1250, wave32. Matrix ops via V_WMMA_F32_16X16X4_F32 (fp32 ref -> fp32 WMMA).
// Staging via GLOBAL_LOAD_ASYNC_TO_LDS_B128 (ASYNCcnt) where layout permits.

typedef __attribute__((ext_vector_type(2))) float v2f;
typedef __attribute__((ext_vector_type(8))) float v8f;
typedef int v4i_b __attribute__((vector_size(4 * sizeof(int))));  // matches builtin param type

#define AS1 __attribute__((address_space(1)))
#define AS3 __attribute__((address_space(3)))

#define TTS_DIM   1024
#define LLAMA_DIM 3072
#define NMEL      100
#define SEQ_T     1024
#define LN_EPS    1e-5f

__device__ __forceinline__ v8f wmma_f32(v2f a, v2f b, v8f c) {
#if __has_builtin(__builtin_amdgcn_wmma_f32_16x16x4_f32)
  // 8 args: (neg_a, A, neg_b, B, c_mod, C, reuse_a, reuse_b)
  return __builtin_amdgcn_wmma_f32_16x16x4_f32(false, a, false, b, (short)0, c, false, false);
#else
  c[0] += a.x * b.x + a.y * b.y;   // compile-safe fallback, never taken on gfx1250
  return c;
#endif
}

#if __has_builtin(__builtin_amdgcn_global_load_async_to_lds_b128)
#define HAVE_ASYNC_LDS 1
__device__ __forceinline__ void async_copy_b128(const float* g, float* l) {
  __builtin_amdgcn_global_load_async_to_lds_b128(
      (AS1 v4i_b*)const_cast<float*>(g), (AS3 v4i_b*)l, 0, 0);
}
__device__ __forceinline__ void async_wait0() {
#if __has_builtin(__builtin_amdgcn_s_wait_asynccnt)
  __builtin_amdgcn_s_wait_asynccnt(0);
#else
  asm volatile("s_wait_asynccnt 0x0" ::: "memory");
#endif
}
#else
#define HAVE_ASYNC_LDS 0
__device__ __forceinline__ void async_copy_b128(const float*, float*) {}
__device__ __forceinline__ void async_wait0() {}
#endif

// ---------------------------------------------------------------------------
// fp32 WMMA GEMM:  C[M,N] = op(A)[M,K] * op(B)[K,N]  (+ epilogue)
//   TA=false: A row-major [M,K] (lda=K)   TA=true: A stored [K,M] (lda=M)
//   TB=false: B row-major [K,N] (ldb=N)   TB=true: B stored [N,K] (ldb=K)
//   GUARDB : bounds-check B rows against Nreal (only needed for padded-N mel)
//   EPI: 0 plain / 1 col-bias ep[N] / 2 *=sigmoid(ep[m*ldc+n]) / 3 mel store
// Block tile 128x64, 256 threads = 8 waves (4 along M x 2 along N), each wave
// owns a 32x32 sub-tile (4 accumulators). LDS tiles fragment-major ([m][k],
// [n][k], 20-float padded rows -> each operand pair is one aligned
// ds_load_b64, conflict-free banking). Double-buffered: one barrier per
// 16-wide K stage. Staging uses async Global->LDS b128 when the copy is
// contiguous (!TA && TB && !GUARDB), else a register path (overlapped via
// prefetch regs). M mult of 128, N mult of 64 (padded), K mult of 16.
// ---------------------------------------------------------------------------
template<bool TA, bool TB, bool GUARDB, int EPI>
__global__ __launch_bounds__(256)
void gemm_wmma(const float* __restrict__ A, const float* __restrict__ B,
               float* __restrict__ C, const float* __restrict__ ep,
               int M, int N, int K, int lda, int ldb, int ldc, int Nreal)
{
  __shared__ __align__(16) float As[2][128][20];  // [m][k] rows, 80B stride
  __shared__ __align__(16) float Bs[2][64][20];   // [n][k] rows

  constexpr bool USE_ASYNC = (HAVE_ASYNC_LDS != 0) && !TA && TB && !GUARDB;

  const int tid   = threadIdx.x;
  const int lane  = tid & 31;
  const int wave  = tid >> 5;
  const int half  = lane >> 4;        // 0: lanes 0-15, 1: lanes 16-31
  const int l15   = lane & 15;
  const int m_off = (wave & 3) << 5;  // 4 waves along M, 32 rows each
  const int n_off = (wave >> 2) << 5; // 2 waves along N, 32 cols each
  const int blockM = blockIdx.y * 128;
  const int blockN = blockIdx.x * 64;

  v8f c00 = {}, c01 = {}, c10 = {}, c11 = {};

  float4 ra0, ra1, rb0;               // register-path prefetch

  // Issue all staging for K-slab k0 into LDS buffer p (async path), or fetch
  // into registers (register path).
  auto issue_async = [&](int k0, int p) {
    const int m_l = tid >> 2;
    const int k_l = (tid & 3) << 2;
    async_copy_b128(A + (size_t)(blockM + m_l)      * lda + (k0 + k_l), &As[p][m_l][k_l]);
    async_copy_b128(A + (size_t)(blockM + 64 + m_l) * lda + (k0 + k_l), &As[p][64 + m_l][k_l]);
    async_copy_b128(B + (size_t)(blockN + m_l)      * ldb + (k0 + k_l), &Bs[p][m_l][k_l]);
  };

  auto fetch = [&](int k0) {
    if (!TA) {
      const int m_l = tid >> 2;
      const int k_l = (tid & 3) << 2;
      ra0 = *(const float4*)(A + (size_t)(blockM + m_l)      * lda + (k0 + k_l));
      ra1 = *(const float4*)(A + (size_t)(blockM + 64 + m_l) * lda + (k0 + k_l));
    } else {
      const int k_l = tid >> 4;
      const int m_l = (tid & 15) << 3;
      ra0 = *(const float4*)(A + (size_t)(k0 + k_l) * lda + (blockM + m_l));
      ra1 = *(const float4*)(A + (size_t)(k0 + k_l) * lda + (blockM + m_l + 4));
    }
    if (TB) {
      const int n_l = tid >> 2;
      const int k_l = (tid & 3) << 2;
      const int n_g = blockN + n_l;
      if (GUARDB) {
        rb0 = make_float4(0.f, 0.f, 0.f, 0.f);
        if (n_g < Nreal)
          rb0 = *(const float4*)(B + (size_t)n_g * ldb + (k0 + k_l));
      } else {
        rb0 = *(const float4*)(B + (size_t)n_g * ldb + (k0 + k_l));
      }
    } else {
      const int k_l = tid >> 4;
      const int n_l = (tid & 15) << 2;
      const int n_g = blockN + n_l;
      if (!GUARDB || n_g + 3 < Nreal) {
        rb0 = *(const float4*)(B + (size_t)(k0 + k_l) * ldb + n_g);
      } else {
        rb0.x = (n_g + 0 < Nreal) ? B[(size_t)(k0 + k_l) * ldb + n_g + 0] : 0.f;
        rb0.y = (n_g + 1 < Nreal) ? B[(size_t)(k0 + k_l) * ldb + n_g + 1] : 0.f;
        rb0.z = (n_g + 2 < Nreal) ? B[(size_t)(k0 + k_l) * ldb + n_g + 2] : 0.f;
        rb0.w = (n_g + 3 < Nreal) ? B[(size_t)(k0 + k_l) * ldb + n_g + 3] : 0.f;
      }
    }
  };

  auto stage = [&](int p) {
    if (!TA) {
      const int m_l = tid >> 2;
      const int k_l = (tid & 3) << 2;
      *(float4*)&As[p][m_l][k_l]      = ra0;
      *(float4*)&As[p][64 + m_l][k_l] = ra1;
    } else {
      const int k_l = tid >> 4;
      const int m_l = (tid & 15) << 3;
      As[p][m_l + 0][k_l] = ra0.x; As[p][m_l + 1][k_l] = ra0.y;
      As[p][m_l + 2][k_l] = ra0.z; As[p][m_l + 3][k_l] = ra0.w;
      As[p][m_l + 4][k_l] = ra1.x; As[p][m_l + 5][k_l] = ra1.y;
      As[p][m_l + 6][k_l] = ra1.z; As[p][m_l + 7][k_l] = ra1.w;
    }
    if (TB) {
      const int n_l = tid >> 2;
      const int k_l = (tid & 3) << 2;
      *(float4*)&Bs[p][n_l][k_l] = rb0;
    } else {
      const int k_l = tid >> 4;
      const int n_l = (tid & 15) << 2;
      Bs[p][n_l + 0][k_l] = rb0.x; Bs[p][n_l + 1][k_l] = rb0.y;
      Bs[p][n_l + 2][k_l] = rb0.z; Bs[p][n_l + 3][k_l] = rb0.w;
    }
  };

  const int nst = K >> 4;
  if (USE_ASYNC) {
    issue_async(0, 0);
    async_wait0();
  } else {
    fetch(0);
    stage(0);
  }
  __syncthreads();

  for (int s = 0; s < nst; ++s) {
    const int p = s & 1;
    if (s + 1 < nst) {
      if (USE_ASYNC) issue_async((s + 1) << 4, p ^ 1);
      else           fetch((s + 1) << 4);
    }

#pragma unroll
    for (int kk = 0; kk < 16; kk += 4) {
      const int kb = kk + 2 * half;  // lanes 0-15: K=kk,kk+1 ; 16-31: K=kk+2,kk+3
      const v2f a0 = *(const v2f*)&As[p][m_off + l15][kb];
      const v2f a1 = *(const v2f*)&As[p][m_off + 16 + l15][kb];
      const v2f b0 = *(const v2f*)&Bs[p][n_off + l15][kb];
      const v2f b1 = *(const v2f*)&Bs[p][n_off + 16 + l15][kb];
      c00 = wmma_f32(a0, b0, c00);
      c01 = wmma_f32(a0, b1, c01);
      c10 = wmma_f32(a1, b0, c10);
      c11 = wmma_f32(a1, b1, c11);
    }

    if (s + 1 < nst) {
      if (USE_ASYNC) async_wait0();
      else           stage(p ^ 1);
    }
    __syncthreads();
  }

  // ---- epilogue ----
#pragma unroll
  for (int mi = 0; mi < 2; ++mi) {
#pragma unroll
    for (int nj = 0; nj < 2; ++nj) {
      const v8f& cc = (mi == 0) ? ((nj == 0) ? c00 : c01)
                                : ((nj == 0) ? c10 : c11);
      const int ng = blockN + n_off + 16 * nj + l15;
#pragma unroll
      for (int r = 0; r < 8; ++r) {
        const int mg = blockM + m_off + 16 * mi + r + 8 * half;
        const float val = cc[r];
        if (EPI == 0) {
          C[(size_t)mg * ldc + ng] = val;
        } else if (EPI == 1) {
          C[(size_t)mg * ldc + ng] = val + ep[ng];
        } else if (EPI == 2) {
          const float sg = 1.f / (1.f + expf(-ep[(size_t)mg * ldc + ng]));
          C[(size_t)mg * ldc + ng] = val * sg;
        } else { // EPI == 3 : mel projection, transposed store [B, NMEL, T]
          if (ng < NMEL) {
            const int b = mg >> 10;           // SEQ_T == 1024
            const int t = mg & (SEQ_T - 1);
            C[((size_t)b * NMEL + ng) * SEQ_T + t] = val + ep[ng];
          }
        }
      }
    }
  }
  (void)M; (void)N; (void)Nreal;
}

// ---------------------------------------------------------------------------
// Grouped Conv1d (C=1024, groups=16 -> 64 ch/group, K=3, pad=1) over [B,T,D]
// channel-contiguous layout. One block handles 8 timesteps of one batch so
// each weight load is reused 8x. Optional exact GELU.
// ---------------------------------------------------------------------------
__global__ __launch_bounds__(256)
void grouped_conv_k(const float* __restrict__ x, const float* __restrict__ w,
                    const float* __restrict__ bias, float* __restrict__ y,
                    int applyGelu)
{
  __shared__ float tile[10][TTS_DIM];            // rows t0-1 .. t0+8
  const int tid = threadIdx.x;
  const int b   = blockIdx.x >> 7;               // 128 chunks of 8 per batch
  const int t0  = (blockIdx.x & 127) << 3;

  for (int r = 0; r < 10; ++r) {
    const int tt = t0 + r - 1;
    const bool ok = (tt >= 0) && (tt < SEQ_T);
    const float* src = x + ((size_t)b * SEQ_T + tt) * TTS_DIM;
    for (int d = tid; d < TTS_DIM; d += 256)
      tile[r][d] = ok ? src[d] : 0.f;
  }
  __syncthreads();

  for (int cidx = 0; cidx < 4; ++cidx) {
    const int c    = tid + (cidx << 8);
    const int base = (c >> 6) << 6;              // group input-channel base
    const float* wc = w + (size_t)c * 192;       // [64][3] weights for channel c
    const float bv = bias[c];
    float acc[8];
#pragma unroll
    for (int j = 0; j < 8; ++j) acc[j] = bv;
    for (int ci = 0; ci < 64; ++ci) {
      const float w0 = wc[3 * ci + 0];
      const float w1 = wc[3 * ci + 1];
      const float w2 = wc[3 * ci + 2];
      const int d = base + ci;
#pragma unroll
      for (int j = 0; j < 8; ++j)
        acc[j] = fmaf(w0, tile[j][d],
                 fmaf(w1, tile[j + 1][d],
                 fmaf(w2, tile[j + 2][d], acc[j])));
    }
#pragma unroll
    for (int j = 0; j < 8; ++j) {
      float v = acc[j];
      if (applyGelu)
        v = 0.5f * v * (1.f + erff(v * 0.70710678118654752f));  // exact GELU
      y[((size_t)b * SEQ_T + (t0 + j)) * TTS_DIM + c] = v;
    }
  }
}

// ---------------------------------------------------------------------------
// LayerNorm over last dim (D=1024); one block (256 thr) per row.
// ---------------------------------------------------------------------------
__global__ __launch_bounds__(256)
void layernorm_k(const float* __restrict__ x, const float* __restrict__ g,
                 const float* __restrict__ bta, float* __restrict__ y)
{
  __shared__ float red[256];
  __shared__ float s_mu, s_rstd;
  const int tid = threadIdx.x;
  const float* xr = x + (size_t)blockIdx.x * TTS_DIM;

  const float4 v = *(const float4*)(xr + tid * 4);
  red[tid] = v.x + v.y + v.z + v.w;
  __syncthreads();
  for (int off = 128; off > 0; off >>= 1) {
    if (tid < off) red[tid] += red[tid + off];
    __syncthreads();
  }
  if (tid == 0) s_mu = red[0] * (1.f / (float)TTS_DIM);
  __syncthreads();
  const float mu = s_mu;
  const float d0 = v.x - mu, d1 = v.y - mu, d2 = v.z - mu, d3 = v.w - mu;
  red[tid] = d0 * d0 + d1 * d1 + d2 * d2 + d3 * d3;
  __syncthreads();
  for (int off = 128; off > 0; off >>= 1) {
    if (tid < off) red[tid] += red[tid + off];
    __syncthreads();
  }
  if (tid == 0) s_rstd = rsqrtf(red[0] * (1.f / (float)TTS_DIM) + LN_EPS);
  __syncthreads();
  const float r = s_rstd;
  const int c = tid * 4;
  const float4 go = *(const float4*)(g + c);
  const float4 bo = *(const float4*)(bta + c);
  float4 o;
  o.x = d0 * r * go.x + bo.x;
  o.y = d1 * r * go.y + bo.y;
  o.z = d2 * r * go.z + bo.z;
  o.w = d3 * r * go.w + bo.w;
  *(float4*)(y + (size_t)blockIdx.x * TTS_DIM + c) = o;
}

// ---------------------------------------------------------------------------
// Pipeline (timesteps input is unused by the reference math):
//  weights:  Ta = W_in*Wd^T + bd ; Tb = P^T*Ta ; Ta = (Tb*P) o sigmoid(mask) ;
//            Tb = P*Ta ; Ta = Tb*P^T          (= fused operator M)
//  acts:     X = llama*W_in^T + b_in ; F = X*M^T
//            H = gelu(conv1(F)) ; X = conv2(H) ; F = LN(X)
//            out = transpose_store(F*Wmel^T + bmel)
// ---------------------------------------------------------------------------
extern "C" void kernel_launch(void* const* d_in, const int* in_sizes, int n_in,
                              void* d_out, int out_size, void* d_ws, size_t ws_size,
                              hipStream_t stream) {
  const float* llama = (const float*)d_in[0];
  /* d_in[1] = timesteps : unused by reference computation */
  const float* W_in  = (const float*)d_in[2];
  const float* b_in  = (const float*)d_in[3];
  const float* P     = (const float*)d_in[4];
  const float* mask  = (const float*)d_in[5];
  const float* Wd    = (const float*)d_in[6];
  const float* bd    = (const float*)d_in[7];
  const float* c1w   = (const float*)d_in[8];
  const float* c1b   = (const float*)d_in[9];
  const float* c2w   = (const float*)d_in[10];
  const float* c2b   = (const float*)d_in[11];
  const float* lng   = (const float*)d_in[12];
  const float* lnb   = (const float*)d_in[13];
  const float* Wmel  = (const float*)d_in[14];
  const float* bmel  = (const float*)d_in[15];
  float* out = (float*)d_out;

  // Workspace: X(16MB) | F(16MB) | H(16MB, overlaid by Ta/Tb before convs run)
  float* X  = (float*)d_ws;
  float* F  = X + (size_t)4096 * 1024;
  float* H  = F + (size_t)4096 * 1024;
  float* Ta = H;                         // 1024x1024, dead before H is written
  float* Tb = H + (size_t)1024 * 1024;

  const dim3 blk(256);
  const dim3 gW(16, 8);    // 1024x1024: N/64 x M/128
  const dim3 gX(16, 32);   // 4096x1024

  // --- weight-side precompute of the fused CSM operator ---
  gemm_wmma<false, true , false, 1><<<gW, blk, 0, stream>>>(W_in, Wd, Ta, bd,
      1024, 1024, LLAMA_DIM, LLAMA_DIM, LLAMA_DIM, 1024, 1024);          // Ta = W_in*Wd^T + bd
  gemm_wmma<true , false, false, 0><<<gW, blk, 0, stream>>>(P, Ta, Tb, nullptr,
      1024, 1024, 1024, 1024, 1024, 1024, 1024);                         // Tb = P^T * Ta
  gemm_wmma<false, false, false, 2><<<gW, blk, 0, stream>>>(Tb, P, Ta, mask,
      1024, 1024, 1024, 1024, 1024, 1024, 1024);                         // Ta = (Tb*P) o sigmoid(mask)
  gemm_wmma<false, false, false, 0><<<gW, blk, 0, stream>>>(P, Ta, Tb, nullptr,
      1024, 1024, 1024, 1024, 1024, 1024, 1024);                         // Tb = P * Ta
  gemm_wmma<false, true , false, 0><<<gW, blk, 0, stream>>>(Tb, P, Ta, nullptr,
      1024, 1024, 1024, 1024, 1024, 1024, 1024);                         // Ta = Tb * P^T  (= M)

  // --- activation path ---
  gemm_wmma<false, true , false, 1><<<gX, blk, 0, stream>>>(llama, W_in, X, b_in,
      4096, 1024, LLAMA_DIM, LLAMA_DIM, LLAMA_DIM, 1024, 1024);          // X = llama*W_in^T + b_in
  gemm_wmma<false, true , false, 0><<<gX, blk, 0, stream>>>(X, Ta, F, nullptr,
      4096, 1024, 1024, 1024, 1024, 1024, 1024);                         // F = X * M^T

  grouped_conv_k<<<512, blk, 0, stream>>>(F, c1w, c1b, H, 1);            // H = gelu(conv1(F))
  grouped_conv_k<<<512, blk, 0, stream>>>(H, c2w, c2b, X, 0);            // X = conv2(H)
  layernorm_k<<<4096, blk, 0, stream>>>(X, lng, lnb, F);                 // F = LN(X)

  gemm_wmma<false, true , true , 3><<<dim3(2, 32), blk, 0, stream>>>(F, Wmel, out, bmel,
      4096, 112, 1024, 1024, 1024, 0, NMEL);                             // out[B,NMEL,T]

  (void)in_sizes; (void)n_in; (void)out_size; (void)ws_size;
}